// NodeNetwork_69415261438420
// MI455X (gfx1250) — compile-verified
//
#include <hip/hip_runtime.h>

typedef __attribute__((ext_vector_type(2))) float v2f;
typedef __attribute__((ext_vector_type(4))) float f4;
typedef __attribute__((ext_vector_type(8))) float v8f;

#define EHD 128
#define FD  16
#define HD  128
#define DEG 32
#define KX  (EHD + FD + HD)   // 272
#define XSTRIDE 276           // 276%64=20 -> conflict-free A loads, 16B aligned rows
#define HSTRIDE 132           // 132%64=4  -> conflict-free A loads, 16B aligned rows
#define TM  16                // nodes per workgroup

__launch_bounds__(256, 4)
__global__ void node_mlp_kernel(const float* __restrict__ mailbox,
                                const float* __restrict__ feat,
                                const float* __restrict__ hid,
                                const float* __restrict__ W1, const float* __restrict__ b1,
                                const float* __restrict__ W2, const float* __restrict__ b2,
                                const float* __restrict__ W3, const float* __restrict__ b3,
                                float* __restrict__ out, int n_nodes)
{
    __shared__ float xs[TM * XSTRIDE];   // [16][272] concat input, padded
    __shared__ float hA[TM * HSTRIDE];   // [16][128] hidden after layer 1
    __shared__ float hB[TM * HSTRIDE];   // [16][128] hidden after layer 2

    const int tid  = threadIdx.x;
    const int base = blockIdx.x * TM;

    // ---------------- Stage 1: mailbox reduction (bandwidth-critical) ----------------
    {
        const int l32 = tid & 31;        // 32 lanes cover 128 channels as float4
        const int grp = tid >> 5;        // 8 groups, 2 nodes each
        const int c4  = l32 * 4;
        #pragma unroll
        for (int j = 0; j < 2; ++j) {
            int n  = 2 * grp + j;
            int gn = base + n;
            if (gn >= n_nodes) gn = n_nodes - 1;       // clamp (tail-safe loads)
            const f4* mb = (const f4*)(mailbox + ((size_t)gn * DEG) * EHD + c4);
            f4 acc = {0.f, 0.f, 0.f, 0.f};
            #pragma unroll 8
            for (int d = 0; d < DEG; ++d) {
                f4 v = __builtin_nontemporal_load(mb + d * (EHD / 4));
                acc.x += v.x; acc.y += v.y; acc.z += v.z; acc.w += v.w;
            }
            *(f4*)(&xs[n * XSTRIDE + c4]) = acc;
        }
        // concat node_features (cols 128..143) and node_hidden_rep (cols 144..271)
        for (int i = tid; i < TM * (FD + HD); i += 256) {
            int n = i / (FD + HD);
            int c = i - n * (FD + HD);
            int gn = base + n;
            if (gn >= n_nodes) gn = n_nodes - 1;
            float v = (c < FD) ? feat[(size_t)gn * FD + c]
                               : hid[(size_t)gn * HD + (c - FD)];
            xs[n * XSTRIDE + EHD + c] = v;
        }
    }
    __syncthreads();

    // ---------------- WMMA MLP: wave w owns output columns [16w, 16w+16) ----------------
    const int w       = tid >> 5;          // wave id 0..7 = N-tile
    const int lane    = tid & 31;
    const int r       = lane & 15;         // A row / B column within tile
    const int khalf   = (lane >> 4) * 2;   // K sub-offset per ISA f32 A/B layout
    const int col     = w * 16 + r;        // global output column
    const int rowbase = (lane >> 4) * 8;   // C/D row base per ISA layout

    // GEMM1: xs[16x272] @ W1[272x128] + b1, ReLU
    v8f c1 = {};
    for (int k = 0; k < KX; k += 4) {
        int kk = k + khalf;
        v2f a; a.x = xs[r * XSTRIDE + kk];  a.y = xs[r * XSTRIDE + kk + 1];
        v2f b; b.x = W1[kk * HD + col];     b.y = W1[(kk + 1) * HD + col];
        c1 = __builtin_amdgcn_wmma_f32_16x16x4_f32(false, a, false, b,
                                                   (short)0, c1, false, false);
    }
    {
        float bias = b1[col];
        #pragma unroll
        for (int i = 0; i < 8; ++i) {
            float v = c1[i] + bias;
            hA[(rowbase + i) * HSTRIDE + col] = v > 0.f ? v : 0.f;
        }
    }
    __syncthreads();

    // GEMM2: hA[16x128] @ W2[128x128] + b2, ReLU
    v8f c2 = {};
    for (int k = 0; k < HD; k += 4) {
        int kk = k + khalf;
        v2f a; a.x = hA[r * HSTRIDE + kk];  a.y = hA[r * HSTRIDE + kk + 1];
        v2f b; b.x = W2[kk * HD + col];     b.y = W2[(kk + 1) * HD + col];
        c2 = __builtin_amdgcn_wmma_f32_16x16x4_f32(false, a, false, b,
                                                   (short)0, c2, false, false);
    }
    {
        float bias = b2[col];
        #pragma unroll
        for (int i = 0; i < 8; ++i) {
            float v = c2[i] + bias;
            hB[(rowbase + i) * HSTRIDE + col] = v > 0.f ? v : 0.f;
        }
    }
    __syncthreads();

    // GEMM3: hB[16x128] @ W3[128x128] + b3 -> out
    v8f c3 = {};
    for (int k = 0; k < HD; k += 4) {
        int kk = k + khalf;
        v2f a; a.x = hB[r * HSTRIDE + kk];  a.y = hB[r * HSTRIDE + kk + 1];
        v2f b; b.x = W3[kk * HD + col];     b.y = W3[(kk + 1) * HD + col];
        c3 = __builtin_amdgcn_wmma_f32_16x16x4_f32(false, a, false, b,
                                                   (short)0, c3, false, false);
    }
    {
        float bias = b3[col];
        #pragma unroll
        for (int i = 0; i < 8; ++i) {
            int node = base + rowbase + i;
            if (node < n_nodes)
                out[(size_t)node * HD + col] = c3[i] + bias;
        }
    }
}

extern "C" void kernel_launch(void* const* d_in, const int* in_sizes, int n_in,
                              void* d_out, int out_size, void* d_ws, size_t ws_size,
                              hipStream_t stream) {
    const float* mailbox = (const float*)d_in[0];
    const float* feat    = (const float*)d_in[1];
    const float* hid     = (const float*)d_in[2];
    const float* W1      = (const float*)d_in[3];
    const float* b1      = (const float*)d_in[4];
    const float* W2      = (const float*)d_in[5];
    const float* b2      = (const float*)d_in[6];
    const float* W3      = (const float*)d_in[7];
    const float* b3      = (const float*)d_in[8];
    float* out = (float*)d_out;

    int n_nodes = in_sizes[1] / FD;               // node_features is [N,16]
    int blocks  = (n_nodes + TM - 1) / TM;
    hipLaunchKernelGGL(node_mlp_kernel, dim3(blocks), dim3(256), 0, stream,
                       mailbox, feat, hid, W1, b1, W2, b2, W3, b3, out, n_nodes);
}